// GraphEncoder_47493748359349
// MI455X (gfx1250) — compile-verified
//
#include <hip/hip_runtime.h>
#include <hip/hip_bf16.h>
#include <stdint.h>

// ---------------------------------------------------------------------------
// 2-layer GCN for MI455X (gfx1250, wave32).
//   layer: out = D^-1/2 (A+I) D^-1/2 (x @ W) + b,  relu between layers.
// - GEMM: native fp32 WMMA (v_wmma_f32_16x16x4_f32), double-buffered LDS
//   staging via global_load_async_to_lds_b128 + s_wait_asynccnt.
// - Aggregation: CSR gather (counting sort by target) -> zero float atomics;
//   all gathered traffic is L2-resident (xw = 51.2 MB << 192 MB L2).
// ---------------------------------------------------------------------------

typedef float v2f __attribute__((ext_vector_type(2)));
typedef float v8f __attribute__((ext_vector_type(8)));
typedef int   v4i __attribute__((vector_size(16)));   // matches builtin proto

#define NNODES 50000
#define FDIM   256
#define KC     16      // K-chunk staged per LDS buffer
#define LDA_PAD 20     // LDS dwords per A row (bank-conflict-free, 16B-aligned)

#if __has_builtin(__builtin_amdgcn_global_load_async_to_lds_b128)
#define HAVE_ASYNC_LDS 1
#endif

typedef __attribute__((address_space(1))) v4i g_v4i;   // global int4
typedef __attribute__((address_space(3))) v4i l_v4i;   // LDS int4

__device__ __forceinline__ void async_copy16(const float* g, float* l) {
#if defined(HAVE_ASYNC_LDS)
    __builtin_amdgcn_global_load_async_to_lds_b128(
        (g_v4i*)(uintptr_t)g,
        (l_v4i*)(unsigned)(uintptr_t)l, 0, 0);
#else
    // CDNA5 ISA: GLOBAL_LOAD_ASYNC_TO_LDS_B128 (GV mode)
    asm volatile("global_load_async_to_lds_b128 %0, %1, off"
                 :: "v"((unsigned)(uintptr_t)l), "v"(g) : "memory");
#endif
}

__device__ __forceinline__ void wait_async0() {
#if defined(HAVE_ASYNC_LDS) && __has_builtin(__builtin_amdgcn_s_wait_asynccnt)
    __builtin_amdgcn_s_wait_asynccnt(0);
#else
    asm volatile("s_wait_asynccnt 0x0" ::: "memory");
#endif
}

// ---------------- CSR build: counts -> scan -> fill ----------------

__global__ void gcn_zero_i32(int* __restrict__ p, int n) {
    int i = blockIdx.x * blockDim.x + threadIdx.x;
    if (i < n) p[i] = 0;
}

__global__ void gcn_count(const long long* __restrict__ col,
                          int* __restrict__ counts, long long E) {
    long long e = (long long)blockIdx.x * blockDim.x + threadIdx.x;
    if (e < E) atomicAdd(&counts[(int)col[e]], 1);
}

// dis[i] = rsqrt(deg) with deg = in-degree + 1 (self loop); deg >= 1 always.
__global__ void gcn_dis(const int* __restrict__ counts,
                        float* __restrict__ dis, int n) {
    int i = blockIdx.x * blockDim.x + threadIdx.x;
    if (i < n) dis[i] = rsqrtf((float)counts[i] + 1.0f);
}

// Single-block exclusive scan over n counts -> offsets[0..n] (n small: 50000).
__global__ __launch_bounds__(1024)
void gcn_scan(const int* __restrict__ counts, int* __restrict__ offsets, int n) {
    __shared__ int buf[1024];
    __shared__ int carry;
    const int tid = threadIdx.x;
    if (tid == 0) carry = 0;
    __syncthreads();
    for (int base = 0; base < n; base += 1024) {
        int i = base + tid;
        int v = (i < n) ? counts[i] : 0;
        buf[tid] = v;
        __syncthreads();
        #pragma unroll
        for (int off = 1; off < 1024; off <<= 1) {
            int t = (tid >= off) ? buf[tid - off] : 0;
            __syncthreads();
            buf[tid] += t;
            __syncthreads();
        }
        int incl = buf[tid];
        int base_c = carry;
        if (i < n) offsets[i] = base_c + (incl - v);
        __syncthreads();                 // everyone has read carry
        if (tid == 1023) carry = base_c + incl;
        __syncthreads();                 // update visible before next round
    }
    if (tid == 0) offsets[n] = carry;
}

__global__ void gcn_fill(const long long* __restrict__ ei,   // [2,E]
                         int* __restrict__ cursor,
                         int* __restrict__ srcIdx, long long E) {
    long long e = (long long)blockIdx.x * blockDim.x + threadIdx.x;
    if (e >= E) return;
    int row = (int)ei[e];
    int col = (int)ei[E + e];
    int pos = atomicAdd(&cursor[col], 1);
    srcIdx[pos] = row;
}

// ---------------- fp32 WMMA GEMM: C[M,256] = op(A)[M,K] @ B[K,256] ----------
// One block per 16-row strip; 8 waves x two 16x16 tiles cover N=256.
// K staged through double-buffered LDS with async global->LDS copies.

template <bool RELU_IN>
__global__ __launch_bounds__(256)
void gcn_gemm_wmma_f32(const float* __restrict__ A,
                       const float* __restrict__ B,
                       float* __restrict__ C, int M, int K, int N) {
    __shared__ float ldsA[2][16 * LDA_PAD];   // 16 x KC, padded rows
    __shared__ float ldsB[2][KC * 256];       // KC x 256

    const int tid  = threadIdx.x;
    const int lane = tid & 31;
    const int wave = tid >> 5;
    const int tm   = blockIdx.x;              // 16-row strip index
    const int half = lane >> 4;               // K sub-pair select
    const int r    = lane & 15;               // M index (A) / N index (B, C)
    const int tn0  = wave * 2;                // this wave's two 16-col tiles

    const float* Ablk = A + (size_t)tm * 16 * K;

    auto issue = [&](int c, int b) {
        const int kBase = c * KC;
        // B chunk: KC x 256 floats = 1024 x b128, 4 per thread.
        #pragma unroll
        for (int i = 0; i < 4; ++i) {
            int idx  = tid + i * 256;
            int krow = idx >> 6;
            int seg  = idx & 63;
            async_copy16(B + (size_t)(kBase + krow) * N + seg * 4,
                         &ldsB[b][krow * 256 + seg * 4]);
        }
        // A chunk: 16 x KC floats = 64 x b128, threads 0..63.
        if (tid < 64) {
            int row = tid >> 2;
            int seg = tid & 3;
            async_copy16(Ablk + (size_t)row * K + kBase + seg * 4,
                         &ldsA[b][row * LDA_PAD + seg * 4]);
        }
    };

    v8f acc0 = {}, acc1 = {};
    const int chunks = K / KC;
    issue(0, 0);
    for (int c = 0; c < chunks; ++c) {
        const int b = c & 1;
        wait_async0();                    // own async copies landed in LDS
        __syncthreads();                  // => everyone's copies landed
        if (c + 1 < chunks) issue(c + 1, b ^ 1);   // overlap with compute
        const float* lA = ldsA[b];
        const float* lB = ldsB[b];
        #pragma unroll
        for (int kk = 0; kk < KC; kk += 4) {
            const int kr = kk + 2 * half;
            v2f a;
            const float2 av = *(const float2*)&lA[r * LDA_PAD + kr]; // ds_load_b64
            a.x = av.x; a.y = av.y;
            if (RELU_IN) { a.x = fmaxf(a.x, 0.0f); a.y = fmaxf(a.y, 0.0f); }
            v2f b0, b1;
            b0.x = lB[(size_t)kr * 256 + tn0 * 16 + r];
            b0.y = lB[(size_t)(kr + 1) * 256 + tn0 * 16 + r];
            b1.x = lB[(size_t)kr * 256 + (tn0 + 1) * 16 + r];
            b1.y = lB[(size_t)(kr + 1) * 256 + (tn0 + 1) * 16 + r];
            acc0 = __builtin_amdgcn_wmma_f32_16x16x4_f32(
                       false, a, false, b0, (short)0, acc0, false, false);
            acc1 = __builtin_amdgcn_wmma_f32_16x16x4_f32(
                       false, a, false, b1, (short)0, acc1, false, false);
        }
        __syncthreads();                  // done reading buf b before overwrite
    }

    // D layout: VGPR v -> M = v + 8*half, N = r
    #pragma unroll
    for (int v = 0; v < 8; ++v) {
        const size_t mrow = (size_t)(tm * 16 + v + 8 * half);
        C[mrow * N + tn0 * 16 + r]       = acc0[v];
        C[mrow * N + (tn0 + 1) * 16 + r] = acc1[v];
    }
}

// ---------------- CSR aggregation: one wave per node, no atomics ------------
// out[i] = bias + dis_i^2 * xw[i] + sum_{j in in(i)} dis_i*dis_row * xw[row]

__global__ __launch_bounds__(256)
void gcn_agg_gather(const float* __restrict__ xw,
                    const int* __restrict__ srcIdx,
                    const int* __restrict__ offsets,
                    const float* __restrict__ dis,
                    const float* __restrict__ bias,
                    float* __restrict__ out, int n) {
    const int node = blockIdx.x * (blockDim.x >> 5) + (threadIdx.x >> 5);
    if (node >= n) return;
    const int lane = threadIdx.x & 31;   // lane covers float4 f4=lane and 32+lane

    const int beg = offsets[node], end = offsets[node + 1];
    const float sc = dis[node];

    float ax = 0, ay = 0, az = 0, aw = 0;
    float bx = 0, by = 0, bz = 0, bw = 0;
    for (int j = beg; j < end; ++j) {
        const int row = srcIdx[j];                  // uniform across wave
        const float nrm = sc * dis[row];
        const float4* xr = (const float4*)(xw + (size_t)row * FDIM);
        const float4 p = xr[lane];                  // coalesced 512B
        const float4 q = xr[32 + lane];             // coalesced 512B
        ax += nrm * p.x; ay += nrm * p.y; az += nrm * p.z; aw += nrm * p.w;
        bx += nrm * q.x; by += nrm * q.y; bz += nrm * q.z; bw += nrm * q.w;
    }

    const float nn = sc * sc;                       // self-loop norm
    const float4* xs = (const float4*)(xw + (size_t)node * FDIM);
    const float4 p = xs[lane], q = xs[32 + lane];
    const float4* bb = (const float4*)bias;
    const float4 c0 = bb[lane], c1 = bb[32 + lane];

    float4 o0, o1;
    o0.x = c0.x + nn * p.x + ax;  o0.y = c0.y + nn * p.y + ay;
    o0.z = c0.z + nn * p.z + az;  o0.w = c0.w + nn * p.w + aw;
    o1.x = c1.x + nn * q.x + bx;  o1.y = c1.y + nn * q.y + by;
    o1.z = c1.z + nn * q.z + bz;  o1.w = c1.w + nn * q.w + bw;

    float4* od = (float4*)(out + (size_t)node * FDIM);
    od[lane] = o0;
    od[32 + lane] = o1;
}

// ---------------- launch ----------------

extern "C" void kernel_launch(void* const* d_in, const int* in_sizes, int n_in,
                              void* d_out, int out_size, void* d_ws, size_t ws_size,
                              hipStream_t stream) {
    const float*     x  = (const float*)d_in[0];
    const long long* ei = (const long long*)d_in[1];
    const float*     W1 = (const float*)d_in[2];
    const float*     b1 = (const float*)d_in[3];
    const float*     W2 = (const float*)d_in[4];
    const float*     b2 = (const float*)d_in[5];
    float*           out = (float*)d_out;

    const int n = NNODES;
    const long long E = (long long)in_sizes[1] / 2;
    const int IN = 128;
    (void)n_in; (void)out_size; (void)ws_size;

    // workspace layout (all 256B-aligned)
    char* ws = (char*)d_ws;
    size_t off = 0;
    auto alloc = [&](size_t bytes) {
        void* p = ws + off;
        off += (bytes + 255) & ~(size_t)255;
        return p;
    };
    int*   counts  = (int*)  alloc((size_t)n * 4);
    int*   offsets = (int*)  alloc((size_t)(n + 1) * 4);
    int*   cursor  = (int*)  alloc((size_t)n * 4);
    int*   srcIdx  = (int*)  alloc((size_t)E * 4);
    float* dis     = (float*)alloc((size_t)n * 4);
    float* bufA    = (float*)alloc((size_t)n * FDIM * 4);
    float* bufB    = (float*)alloc((size_t)n * FDIM * 4);

    const int nB   = (n + 255) / 256;
    const int eB   = (int)((E + 255) / 256);

    // --- CSR build + normalization (shared by both layers) ---
    gcn_zero_i32<<<nB, 256, 0, stream>>>(counts, n);
    gcn_count   <<<eB, 256, 0, stream>>>(ei + E, counts, E);
    gcn_dis     <<<nB, 256, 0, stream>>>(counts, dis, n);
    gcn_scan    <<<1, 1024, 0, stream>>>(counts, offsets, n);
    (void)hipMemcpyAsync(cursor, offsets, (size_t)n * 4,
                         hipMemcpyDeviceToDevice, stream);
    gcn_fill    <<<eB, 256, 0, stream>>>(ei, cursor, srcIdx, E);

    const int gemmBlocks = n / 16;                 // 50000 % 16 == 0
    const int aggBlocks  = (n + 7) / 8;            // 8 waves (nodes) per block

    // --- layer 1: xw1 = x @ W1 ; h = agg(xw1)+b1 (relu deferred to GEMM2) ---
    gcn_gemm_wmma_f32<false><<<gemmBlocks, 256, 0, stream>>>(x, W1, bufA, n, IN, FDIM);
    gcn_agg_gather<<<aggBlocks, 256, 0, stream>>>(bufA, srcIdx, offsets, dis, b1, bufB, n);

    // --- layer 2: xw2 = relu(h) @ W2 ; out = agg(xw2)+b2 ---
    gcn_gemm_wmma_f32<true><<<gemmBlocks, 256, 0, stream>>>(bufB, W2, bufA, n, FDIM, FDIM);
    gcn_agg_gather<<<aggBlocks, 256, 0, stream>>>(bufA, srcIdx, offsets, dis, b2, out, n);
}